// INGPTable_8057358647426
// MI455X (gfx1250) — compile-verified
//
#include <hip/hip_runtime.h>
#include <hip/hip_bf16.h>

// InstantNGP 2D hash-grid encoding, single level.
//   x:     [B, 2] float32 coords in [0,1)
//   table: [524288, 2] float32
//   out:   [B, 2] float32
//
// Memory-bound: x/out are pure streams (NT cache hints keep them from
// evicting the 4MB table, which is L2-resident on the 192MB L2), table
// gathers use default (RT) policy so they hit L2.

#define NGP_RESOLUTION 1024.0f
#define NGP_TABLE_MASK 524287u      // TABLE_SIZE = 2^19
#define NGP_PI2        2654435761u

typedef float v4f __attribute__((ext_vector_type(4)));
typedef float v2f __attribute__((ext_vector_type(2)));

__global__ __launch_bounds__(256) void ingp_hashgrid_kernel(
    const float* __restrict__ x,
    const float* __restrict__ table,
    float* __restrict__ out,
    int batch)
{
    const int tid  = blockIdx.x * blockDim.x + threadIdx.x;
    const int base = tid * 2;                 // two points per thread
    if (base >= batch) return;

    // Streaming load of 2 points' coords: b128, non-temporal.
    const v4f xin = __builtin_nontemporal_load(
        reinterpret_cast<const v4f*>(x) + tid);

    const v2f* __restrict__ tab = reinterpret_cast<const v2f*>(table);

    v4f result;

#pragma unroll
    for (int p = 0; p < 2; ++p) {
        const float xs = xin[2 * p + 0] * NGP_RESOLUTION;
        const float ys = xin[2 * p + 1] * NGP_RESOLUTION;

        const float xf = floorf(xs);
        const float yf = floorf(ys);
        const unsigned c0 = (unsigned)(int)xf;
        const unsigned c1 = (unsigned)(int)yf;

        // spatial hash: (1*c0) ^ (2654435761*c1) mod 2^19  (uint32 == int64 here)
        const unsigned hy0 = c1 * NGP_PI2;
        const unsigned hy1 = (c1 + 1u) * NGP_PI2;
        const unsigned h00 = ( c0        ^ hy0) & NGP_TABLE_MASK;  // (0,0)
        const unsigned h01 = ( c0        ^ hy1) & NGP_TABLE_MASK;  // (0,1)
        const unsigned h10 = ((c0 + 1u)  ^ hy0) & NGP_TABLE_MASK;  // (1,0)
        const unsigned h11 = ((c0 + 1u)  ^ hy1) & NGP_TABLE_MASK;  // (1,1)

        // Table gathers: default temporal policy -> stay L2-resident (4MB table).
        const v2f f00 = tab[h00];
        const v2f f01 = tab[h01];
        const v2f f10 = tab[h10];
        const v2f f11 = tab[h11];

        // Bilinear weights, factored exactly like the reference:
        //   w_corner = (1 - |xs - cx|) * (1 - |ys - cy|)
        const float wx0 = 1.0f - fabsf(xs - xf);
        const float wx1 = 1.0f - fabsf(xs - (xf + 1.0f));
        const float wy0 = 1.0f - fabsf(ys - yf);
        const float wy1 = 1.0f - fabsf(ys - (yf + 1.0f));

        const float w00 = wx0 * wy0;
        const float w01 = wx0 * wy1;
        const float w10 = wx1 * wy0;
        const float w11 = wx1 * wy1;

        const float o0 = f00[0] * w00 + f01[0] * w01 + f10[0] * w10 + f11[0] * w11;
        const float o1 = f00[1] * w00 + f01[1] * w01 + f10[1] * w10 + f11[1] * w11;

        result[2 * p + 0] = o0;
        result[2 * p + 1] = o1;
    }

    // Streaming store of 2 points' features: b128, non-temporal.
    __builtin_nontemporal_store(result, reinterpret_cast<v4f*>(out) + tid);
}

extern "C" void kernel_launch(void* const* d_in, const int* in_sizes, int n_in,
                              void* d_out, int out_size, void* d_ws, size_t ws_size,
                              hipStream_t stream) {
    const float* x     = (const float*)d_in[0];   // [B, 2]
    const float* table = (const float*)d_in[1];   // [524288, 2]
    float*       out   = (float*)d_out;           // [B, 2]

    const int batch    = in_sizes[0] / 2;         // number of points (4194304)
    const int nthreads = (batch + 1) / 2;         // 2 points per thread
    const int block    = 256;                     // 8 wave32 per block
    const int grid     = (nthreads + block - 1) / block;

    ingp_hashgrid_kernel<<<grid, block, 0, stream>>>(x, table, out, batch);
}